// Linear_89773406421071
// MI455X (gfx1250) — compile-verified
//
#include <hip/hip_runtime.h>

typedef __attribute__((ext_vector_type(2))) float v2f;
typedef __attribute__((ext_vector_type(8))) float v8f;

namespace {

constexpr int   kInDim   = 1152;                 // = out dim
constexpr int   kMul     = 128;
constexpr int   kLdsStrF = 132;                  // W^T row stride in floats
constexpr int   kLdsStrV = 66;                   // ... in v2f units (bank (4n+k)%64: conflict-free b64)
constexpr float kAlpha   = 0.08838834764831845f; // 1/sqrt(128)

// One irrep block:  out[z,w,i] = alpha * sum_u W[u,w] * x[z,u,i]
// GEMM (N*D) x 128 x 128; A row r = z*D + i, K index u (memory stride D floats).
// Each wave owns a 16-row x 128-col output stripe: 8 column tiles held in 64 acc VGPRs,
// A loaded from HBM exactly once, B fragments from LDS-resident W^T.
template <int D, int XOFF, int WOFF, int TILES>
__global__ __launch_bounds__(256) void e3nn_linear_wmma(
    const float* __restrict__ X,   // [N, 1152] features
    const float* __restrict__ W,   // flat weight
    float*       __restrict__ Y)   // [N, 1152] output
{
  extern __shared__ float lds[];   // W^T: lds[n*132 + u] = W[u][n]

  // Cooperative transpose-load of this block's 128x128 weight into LDS.
  for (int e = threadIdx.x; e < kMul * kMul; e += 256) {
    const int u = e >> 7;
    const int n = e & 127;
    lds[n * kLdsStrF + u] = W[WOFF + e];
  }
  __syncthreads();

  const int wave = threadIdx.x >> 5;
  const int lane = threadIdx.x & 31;
  const int l16  = lane & 15;
  const int half = lane >> 4;          // 0: lanes 0-15, 1: lanes 16-31

  const int tile = blockIdx.x * 8 + wave;   // 16-row tile (wave-uniform branch)
  if (tile >= TILES) return;

  const int r0 = tile << 4;

  // A operand: lane holds row M = lane%16; VGPR0/1 hold K = {0,1} (low half) / {2,3} (high).
  const int r = r0 + l16;
  const int z = r / D;
  const int i = r - z * D;
  const float* __restrict__ a_ptr = X + z * kInDim + XOFF + i;   // u-stride = D floats

  // B operand via v2f view of LDS -> ds_load_b64 with immediate offsets.
  const v2f* __restrict__ lds2 =
      (const v2f*)__builtin_assume_aligned((const void*)lds, 8);
  const int bn = l16 * kLdsStrV;       // per-lane column row base (v2f units)

  v8f acc[8] = {};                     // 8 column tiles x (16x16 f32 C) = 64 VGPRs

  const int k_lo = half * 2;           // this half-wave's first K inside each K-step
  // 8 groups of 4 K-steps: issue 8 global f32 + 32 ds b64, then 32 WMMAs.
#pragma unroll 1
  for (int g = 0; g < 8; ++g) {
    v2f a4[4];
    v2f b4[4][8];
#pragma unroll
    for (int j = 0; j < 4; ++j) {
      const int ks = g * 4 + j;
      const int k0 = ks * 4 + k_lo;
      a4[j].x = a_ptr[(k0 + 0) * D];
      a4[j].y = a_ptr[(k0 + 1) * D];
#pragma unroll
      for (int ct = 0; ct < 8; ++ct) {
        b4[j][ct] = lds2[bn + ct * (16 * kLdsStrV) + ks * 2 + half];
      }
    }
#pragma unroll
    for (int j = 0; j < 4; ++j) {
#pragma unroll
      for (int ct = 0; ct < 8; ++ct) {
        acc[ct] = __builtin_amdgcn_wmma_f32_16x16x4_f32(
            false, a4[j], false, b4[j][ct], (short)0, acc[ct], false, false);
      }
    }
  }

  // C/D layout: VGPR v -> row (v + 8*half), col = lane%16.
  int row_base[8];
#pragma unroll
  for (int v = 0; v < 8; ++v) {
    const int rr = r0 + half * 8 + v;
    const int zz = rr / D;
    const int ii = rr - zz * D;
    row_base[v] = zz * kInDim + XOFF + ii;
  }
#pragma unroll
  for (int ct = 0; ct < 8; ++ct) {
    const int w = ct * 16 + l16;
#pragma unroll
    for (int v = 0; v < 8; ++v) {
      Y[row_base[v] + w * D] = acc[ct][v] * kAlpha;
    }
  }
}

} // namespace

extern "C" void kernel_launch(void* const* d_in, const int* in_sizes, int n_in,
                              void* d_out, int out_size, void* d_ws, size_t ws_size,
                              hipStream_t stream) {
  const float* X = (const float*)d_in[0];   // features [100000, 1152] f32
  const float* W = (const float*)d_in[1];   // weight   [49152] f32
  float*       Y = (float*)d_out;           // output   [100000, 1152] f32

  constexpr int kLdsBytes = kMul * kLdsStrF * 4;   // 67,584 B (WGP LDS is 320 KB)

  (void)hipFuncSetAttribute((const void*)&e3nn_linear_wmma<1, 0, 0, 6250>,
                            hipFuncAttributeMaxDynamicSharedMemorySize, kLdsBytes);
  (void)hipFuncSetAttribute((const void*)&e3nn_linear_wmma<3, 128, 16384, 18750>,
                            hipFuncAttributeMaxDynamicSharedMemorySize, kLdsBytes);
  (void)hipFuncSetAttribute((const void*)&e3nn_linear_wmma<5, 512, 32768, 31250>,
                            hipFuncAttributeMaxDynamicSharedMemorySize, kLdsBytes);

  // tiles = N*d/16, workgroups = ceil(tiles/8)
  e3nn_linear_wmma<1,   0,     0,  6250><<< 782, 256, kLdsBytes, stream>>>(X, W, Y);
  e3nn_linear_wmma<3, 128, 16384, 18750><<<2344, 256, kLdsBytes, stream>>>(X, W, Y);
  e3nn_linear_wmma<5, 512, 32768, 31250><<<3907, 256, kLdsBytes, stream>>>(X, W, Y);
}